// EmbeddingDropout_31190052504336
// MI455X (gfx1250) — compile-verified
//
#include <hip/hip_runtime.h>
#include <hip/hip_bf16.h>

// EmbeddingDropout gather: out[t, :] = weight[words[t], :] * mask[words[t]]
// words: [8192] int32, weight: [50257, 1024] f32, mask: [50257] f32.
// Pure bandwidth problem (~65 MB @ 23.3 TB/s ~= 2.8 us floor). No matrix math
// -> WMMA inapplicable; the optimal CDNA5 path is pure data movement:
//  - b128 loads / non-temporal b128 stores (one float4 per thread)
//  - scalar (SGPR, KMcnt) loads for block-uniform index + mask
//  - global_prefetch_b8 (SCOPE_DEV -> GL2) for the next token's row
//  - fast-path/tail split so the hot path has zero per-iteration bounds SALU

typedef float fvec4 __attribute__((ext_vector_type(4)));

#define EMB_DIM 1024
#define ROW_BYTES (EMB_DIM * 4)     // 4 KB per row
#define TPB 256                     // 8 wave32 per block; 1 float4 per thread
#define TOK_PER_BLK 4               // tokens per block -> prefetch distance 1

__global__ __launch_bounds__(TPB) void embed_dropout_gather(
    const int* __restrict__ words,
    const float* __restrict__ weight,
    const float* __restrict__ mask,
    float* __restrict__ out,
    int n_tokens)
{
    const int tid  = threadIdx.x;
    const int base = blockIdx.x * TOK_PER_BLK;

    if (base + TOK_PER_BLK <= n_tokens) {
        // ---- fast path: full tile, no per-iteration bounds checks ----
        int w = words[base];  // uniform -> s_load_b32
#pragma unroll
        for (int i = 0; i < TOK_PER_BLK; ++i) {
            const int t = base + i;

            int wn = 0;
            if (i + 1 < TOK_PER_BLK) {
                wn = words[t + 1];  // uniform s_load; doubles as next iter's w
                // Prefetch next 4 KB row into GL2: 32 lanes x 128 B lines.
                if (tid < 32) {
                    const char* pn = (const char*)weight + (size_t)wn * ROW_BYTES;
                    __builtin_prefetch(pn + (size_t)tid * 128, /*rw=*/0, /*locality=*/1);
                }
            }

            const float m = mask[w];  // uniform s_load_b32

            const fvec4* __restrict__ src = (const fvec4*)(weight + (size_t)w * EMB_DIM);
            fvec4 v = src[tid];        // global_load_b128 (RT: keep table in L2)
            v *= m;                    // 4x v_mul_f32

            fvec4* dst = (fvec4*)(out + (size_t)t * EMB_DIM);
            __builtin_nontemporal_store(v, dst + tid);  // global_store_b128 NT

            w = wn;
        }
    } else {
        // ---- tail path: partial tile ----
        for (int t = base; t < n_tokens; ++t) {
            const int w = words[t];
            const float m = mask[w];
            const fvec4* __restrict__ src = (const fvec4*)(weight + (size_t)w * EMB_DIM);
            fvec4 v = src[tid];
            v *= m;
            fvec4* dst = (fvec4*)(out + (size_t)t * EMB_DIM);
            __builtin_nontemporal_store(v, dst + tid);
        }
    }
}

extern "C" void kernel_launch(void* const* d_in, const int* in_sizes, int n_in,
                              void* d_out, int out_size, void* d_ws, size_t ws_size,
                              hipStream_t stream) {
    const int*   words  = (const int*)  d_in[0];  // [4*2048] int32
    const float* weight = (const float*)d_in[1];  // [50257*1024] f32
    const float* mask   = (const float*)d_in[2];  // [50257] f32 (row keep-scale)
    float*       out    = (float*)d_out;          // [4*2048*1024] f32

    const int n_tokens = in_sizes[0];
    const int blocks   = (n_tokens + TOK_PER_BLK - 1) / TOK_PER_BLK;

    embed_dropout_gather<<<blocks, TPB, 0, stream>>>(words, weight, mask, out, n_tokens);
}